// FuseLinkPrediction_15075335209312
// MI455X (gfx1250) — compile-verified
//
#include <hip/hip_runtime.h>

typedef __attribute__((ext_vector_type(2))) float v2f;
typedef __attribute__((ext_vector_type(8))) float v8f;

#define D_HID 64

// ---------------------------------------------------------------------------
// Phase A: fold the two linear layers.
//   wfold[i] = sum_j W1[i][j] * W2[j]      (i = 0..255)
//   bfold    = sum_j b1[j] * W2[j] + b2
// ---------------------------------------------------------------------------
__global__ __launch_bounds__(256)
void fold_weights_kernel(const float* __restrict__ W1,
                         const float* __restrict__ b1,
                         const float* __restrict__ W2,
                         const float* __restrict__ b2,
                         float* __restrict__ wfold,
                         float* __restrict__ bfold) {
  const int i = threadIdx.x;  // 256 threads, one per W1 row
  float acc = 0.f;
#pragma unroll
  for (int j = 0; j < 16; ++j) acc += W1[i * 16 + j] * W2[j];
  wfold[i] = acc;
  if (i == 0) {
    float t = 0.f;
#pragma unroll
    for (int j = 0; j < 16; ++j) t += b1[j] * W2[j];
    bfold[0] = t + b2[0];
  }
}

// ---------------------------------------------------------------------------
// Phase B: per-node scores via V_WMMA_F32_16X16X4_F32.
//   A tile : 16 nodes x 128 features (K), fed 4 K at a time.
//            ISA f32 A layout: lanes 0-15 hold M=0..15 with K pair {k,k+1},
//            lanes 16-31 hold K pair {k+2,k+3}.
//   B      : 128x16, col 0 = w[0:128], col 1 = w[128:256], cols 2..15 = 0.
//            Built with an UNCONDITIONAL b64 load from a safe pointer plus a
//            multiply-mask (no exec-mask branches in the hot loop).
//   D      : col 0 -> s1[node], col 1 -> s2[node]; scores stored [node][2].
//   One wave per 16-node tile; 8 waves per 256-thread block.
// ---------------------------------------------------------------------------
__global__ __launch_bounds__(256)
void node_scores_kernel(const float* __restrict__ h1,
                        const float* __restrict__ h2,
                        const float* __restrict__ wfold,
                        float* __restrict__ scores,
                        int n_nodes, int n_tiles) {
  const int tile = blockIdx.x * 8 + (threadIdx.x >> 5);
  if (tile >= n_tiles) return;  // wave-uniform: EXEC stays all-1 for WMMA

  const int lane = threadIdx.x & 31;
  const int hi   = lane >> 4;   // 0 -> K pair {k,k+1}, 1 -> K pair {k+2,k+3}
  const int m    = lane & 15;   // A: row M (node) / B: col N

  int node   = tile * 16 + m;
  int node_c = node < n_nodes ? node : n_nodes - 1;  // clamp for safe loads
  const float* r1 = h1 + (size_t)node_c * D_HID;
  const float* r2 = h2 + (size_t)node_c * D_HID;

  // Lanes m<2 read real weight columns; lanes m>=2 read the base (safe) and
  // are zeroed by the multiply-mask. All lanes execute identical code.
  const float  bscale = (m < 2) ? 1.f : 0.f;
  const float* wp     = wfold + ((m < 2) ? m * 128 : 0);

  v8f acc = {};

  // K = 0..63  (features from h1; w index = m*128 + k)
#pragma unroll
  for (int s = 0; s < 16; ++s) {
    const int k = 4 * s + 2 * hi;
    v2f a = *(const v2f*)(r1 + k);
    v2f b = *(const v2f*)(wp + k);
    b *= bscale;
    acc = __builtin_amdgcn_wmma_f32_16x16x4_f32(false, a, false, b,
                                                (short)0, acc, false, false);
  }
  // K = 64..127 (features from h2; w index = m*128 + 64 + k)
#pragma unroll
  for (int s = 0; s < 16; ++s) {
    const int k = 4 * s + 2 * hi;
    v2f a = *(const v2f*)(r2 + k);
    v2f b = *(const v2f*)(wp + 64 + k);
    b *= bscale;
    acc = __builtin_amdgcn_wmma_f32_16x16x4_f32(false, a, false, b,
                                                (short)0, acc, false, false);
  }

  // D layout: VGPR r, lanes 0-15 -> D[r][lane]; lanes 16-31 -> D[r+8][lane-16]
  if (m < 2) {
#pragma unroll
    for (int r = 0; r < 8; ++r) {
      const int mm = tile * 16 + 8 * hi + r;
      if (mm < n_nodes) scores[(size_t)mm * 2 + m] = acc[r];
    }
  }
}

// ---------------------------------------------------------------------------
// Phase C: logits[e] = s1[src] + s2[dst] + bb   (pure gather, HBM-bound)
// edges is int64, shape [2, E] row-major.
// ---------------------------------------------------------------------------
__global__ __launch_bounds__(256)
void edge_logits_kernel(const long long* __restrict__ edges,
                        const float* __restrict__ scores,
                        const float* __restrict__ bfold,
                        float* __restrict__ out,
                        int n_edges) {
  const int e = blockIdx.x * blockDim.x + threadIdx.x;
  if (e >= n_edges) return;
  const int src = (int)edges[e];
  const int dst = (int)edges[(size_t)n_edges + e];
  out[e] = scores[(size_t)src * 2] + scores[(size_t)dst * 2 + 1] + bfold[0];
}

// ---------------------------------------------------------------------------
extern "C" void kernel_launch(void* const* d_in, const int* in_sizes, int n_in,
                              void* d_out, int out_size, void* d_ws, size_t ws_size,
                              hipStream_t stream) {
  const float*      h1    = (const float*)d_in[0];
  const float*      h2    = (const float*)d_in[1];
  const float*      W1    = (const float*)d_in[2];  // [256,16]
  const float*      b1    = (const float*)d_in[3];  // [16]
  const float*      W2    = (const float*)d_in[4];  // [16,1]
  const float*      b2    = (const float*)d_in[5];  // [1]
  const long long*  edges = (const long long*)d_in[6];  // int64 [2,E]
  float*            out   = (float*)d_out;

  const int n_nodes = in_sizes[0] / D_HID;
  const int n_edges = in_sizes[6] / 2;

  // Workspace layout (floats): [0..255] wfold, [256] bfold, [512..] scores[2*N]
  float* wfold  = (float*)d_ws;
  float* bfold  = wfold + 256;
  float* scores = wfold + 512;

  fold_weights_kernel<<<1, 256, 0, stream>>>(W1, b1, W2, b2, wfold, bfold);

  const int n_tiles = (n_nodes + 15) / 16;
  const int blocksB = (n_tiles + 7) / 8;
  node_scores_kernel<<<blocksB, 256, 0, stream>>>(h1, h2, wfold, scores,
                                                  n_nodes, n_tiles);

  const int blocksC = (n_edges + 255) / 256;
  edge_logits_kernel<<<blocksC, 256, 0, stream>>>(edges, scores, bfold, out,
                                                  n_edges);
}